// InvariantPointAttention_38706245272121
// MI455X (gfx1250) — compile-verified
//
#include <hip/hip_runtime.h>
#include <hip/hip_bf16.h>
#include <math.h>

// ---------------------------------------------------------------------------
// Invariant Point Attention for MI455X (gfx1250), FP32 end-to-end using
// V_WMMA_F32_16X16X4_F32. Memory-bound on z (75.5MB, fits in 192MB L2), so
// FP32 WMMA is the right precision/perf point: exact math, no conversion
// traffic, and the f32 16x16x4 fragments partition tiles perfectly across a
// wave32 (2 floats/lane for A and B) -> no intra-wave load redundancy.
// Ragged edges are handled with clamped (always-valid) addresses instead of
// per-iteration zero-selects, so the K-loops contain no v_cndmask.
// ---------------------------------------------------------------------------

typedef __attribute__((ext_vector_type(2))) float v2f;
typedef __attribute__((ext_vector_type(8))) float v8f;

#define NRES   384
#define CS     384
#define CZ     128
#define NHEAD  12
#define CH     16
#define NQP    4
#define NPV    8
#define DCAT   2112   // 192 (v_out) + 288 (vp_out) + 96 (norm) + 1536 (pairwise)
#define OFF_VPOUT 192
#define OFF_NORM  480
#define OFF_PAIR  576

// ---- fragment helpers (per CDNA5 ISA 7.12.2 layouts, wave32) --------------
// A 16x4 (MxK):  lane = m + 16*(koff/2); VGPR0=K(koff), VGPR1=K(koff+1)
// B 4x16 (KxN):  lane = n + 16*(koff/2); VGPR0=K(koff), VGPR1=K(koff+1)
// C/D 16x16:     VGPR r: lanes 0-15 -> M=r, lanes 16-31 -> M=r+8; N=lane&15

__device__ __forceinline__ v8f wmma4(v2f a, v2f b, v8f c) {
    return __builtin_amdgcn_wmma_f32_16x16x4_f32(false, a, false, b,
                                                 (short)0, c, false, false);
}

// ---------------------------------------------------------------------------
// Generic GEMM: C[M,N] = A[M,K] @ B[K,N] + bias[N].  M%16==0, K%4==0 assumed.
// N may be ragged (bias projection has N=12): OOB lanes load a clamped
// (duplicate) column — garbage lands only in C columns masked at store.
// 4 waves/block, one 16x16 tile per wave.
// ---------------------------------------------------------------------------
__global__ void gemm_f32_wmma(const float* __restrict__ A, const float* __restrict__ B,
                              const float* __restrict__ bias, float* __restrict__ C,
                              int M, int N, int K, int lda, int ldb, int ldc)
{
    const int lane = threadIdx.x & 31;
    const int wid  = threadIdx.x >> 5;
    const int tiles_n = (N + 15) >> 4;
    const int tiles_m = M >> 4;
    const long tile = (long)blockIdx.x * 4 + wid;
    if (tile >= (long)tiles_m * tiles_n) return;       // whole-wave uniform
    const int m0 = (int)(tile / tiles_n) << 4;
    const int n0 = (int)(tile % tiles_n) << 4;

    const int frag = lane & 15;             // m for A, n for B/C
    const int koff = (lane >> 4) << 1;      // 0 or 2
    const int nb   = n0 + frag;
    const int nbc  = nb < N ? nb : (N - 1); // clamped, always valid

    const float* Arow = A + (long)(m0 + frag) * lda + koff;
    const float* Bcol = B + (long)koff * ldb + nbc;

    v8f acc = {};
    for (int k0 = 0; k0 < K; k0 += 4) {
        v2f a, b;
        a.x = Arow[k0];
        a.y = Arow[k0 + 1];
        b.x = Bcol[(long)k0 * ldb];
        b.y = Bcol[(long)(k0 + 1) * ldb];
        acc = wmma4(a, b, acc);
    }
    if (nb < N) {
        const float bv = bias ? bias[nb] : 0.f;
        const int mbase = (lane >> 4) << 3;
        #pragma unroll
        for (int r = 0; r < 8; ++r)
            C[(long)(m0 + mbase + r) * ldc + nb] = acc[r] + bv;
    }
}

// ---------------------------------------------------------------------------
// Rigid-frame forward warp of projected points.
// proj layout: [i][xyz*NHEAD*P + h*P + p]  ->  out layout: [h][i][p*3+xyz]
// ---------------------------------------------------------------------------
__global__ void warp_points(const float* __restrict__ proj, const float* __restrict__ T,
                            float* __restrict__ out, int P)
{
    int idx = blockIdx.x * blockDim.x + threadIdx.x;   // over NHEAD*P*NRES
    if (idx >= NHEAD * P * NRES) return;
    const int i  = idx % NRES;
    const int hp = idx / NRES;
    const int h = hp / P, p = hp % P;
    const int stride = 3 * NHEAD * P;
    const float* src = proj + (long)i * stride;
    const float x = src[0 * NHEAD * P + h * P + p];
    const float y = src[1 * NHEAD * P + h * P + p];
    const float zc = src[2 * NHEAD * P + h * P + p];
    const float* Ti = T + (long)i * 16;
    float* dst = out + ((long)h * NRES + i) * (3 * P) + p * 3;
    #pragma unroll
    for (int r = 0; r < 3; ++r)
        dst[r] = Ti[r*4+0]*x + Ti[r*4+1]*y + Ti[r*4+2]*zc + Ti[r*4+3];
}

// ---------------------------------------------------------------------------
// Attention logits: wl*(q.k/sqrt(C) + bias - gamma*(wc/2)*sum_p ||tqp-tkp||^2)
// One 16x16 (i,j) tile per wave; q.k via 4x WMMA over C=16.
// grid (24 itile, 6, 12 heads), block 128 -> wave w handles jtile=by*4+w.
// ---------------------------------------------------------------------------
__global__ void ipa_logits(const float* __restrict__ qbuf, const float* __restrict__ kbuf,
                           const float* __restrict__ biasbuf,
                           const float* __restrict__ tqp, const float* __restrict__ tkp,
                           const float* __restrict__ hw, float* __restrict__ att)
{
    const int lane = threadIdx.x & 31;
    const int w = threadIdx.x >> 5;
    const int h = blockIdx.z;
    const int i0 = blockIdx.x << 4;
    const int j0 = (blockIdx.y * 4 + w) << 4;
    const int frag = lane & 15;
    const int koff = (lane >> 4) << 1;

    const float* qrow = qbuf + (long)(i0 + frag) * (NHEAD * CH) + h * CH + koff;
    const float* krow = kbuf + (long)(j0 + frag) * (NHEAD * CH) + h * CH + koff;

    v8f acc = {};
    #pragma unroll
    for (int c0 = 0; c0 < CH; c0 += 4) {
        v2f a, b;
        a.x = qrow[c0]; a.y = qrow[c0 + 1];
        b.x = krow[c0]; b.y = krow[c0 + 1];   // B = k^T: (k=c, n=j)
        acc = wmma4(a, b, acc);
    }

    float gx = hw[h];
    const float gamma = (gx > 20.f) ? gx : log1pf(expf(gx));     // softplus
    const float wc_half = 0.5f * 0.2357022603955158f;            // 0.5*sqrt(1/18)
    const float wl = 0.5773502691896258f;                        // sqrt(1/3)

    const int mbase = (lane >> 4) << 3;
    const int gj = j0 + frag;
    const float* tk = tkp + ((long)h * NRES + gj) * 12;
    float tkl[12];
    #pragma unroll
    for (int t = 0; t < 12; ++t) tkl[t] = tk[t];

    #pragma unroll
    for (int r = 0; r < 8; ++r) {
        const int gi = i0 + mbase + r;
        const float* tq = tqp + ((long)h * NRES + gi) * 12;
        float s = 0.f;
        #pragma unroll
        for (int t = 0; t < 12; ++t) { float d = tq[t] - tkl[t]; s += d * d; }
        const float bia = biasbuf[((long)gi * NRES + gj) * NHEAD + h];
        att[((long)h * NRES + gi) * NRES + gj] =
            wl * (acc[r] * 0.25f + bia - gamma * wc_half * s);
    }
}

// One wave32 per row of 384; 12 elements/lane; shfl_xor reductions.
__global__ void softmax_rows(float* __restrict__ att)
{
    const int lane = threadIdx.x;
    const long base = (long)blockIdx.x * NRES;
    float v[12];
    float mx = -INFINITY;
    #pragma unroll
    for (int t = 0; t < 12; ++t) { v[t] = att[base + lane + t * 32]; mx = fmaxf(mx, v[t]); }
    #pragma unroll
    for (int o = 16; o > 0; o >>= 1) mx = fmaxf(mx, __shfl_xor(mx, o, 32));
    float sum = 0.f;
    #pragma unroll
    for (int t = 0; t < 12; ++t) { v[t] = expf(v[t] - mx); sum += v[t]; }
    #pragma unroll
    for (int o = 16; o > 0; o >>= 1) sum += __shfl_xor(sum, o, 32);
    const float inv = 1.0f / sum;
    #pragma unroll
    for (int t = 0; t < 12; ++t) att[base + lane + t * 32] = v[t] * inv;
}

// pairwise[i,h,cz] = sum_j att[h,i,j] * z[i,j,cz]  (per-i 12x384 @ 384x128)
// grid (384, 2), block 128 -> wave w handles ntile = by*4+w of 8.
// Head rows 12..15 read a clamped duplicate of head 11 (valid memory);
// the resulting garbage C rows are masked at store.
__global__ void pairwise_av(const float* __restrict__ att, const float* __restrict__ z,
                            float* __restrict__ X)
{
    const int lane = threadIdx.x & 31;
    const int w = threadIdx.x >> 5;
    const int i = blockIdx.x;
    const int n0 = (blockIdx.y * 4 + w) << 4;
    const int frag = lane & 15;
    const int koff = (lane >> 4) << 1;
    const int hclamp = frag < NHEAD ? frag : (NHEAD - 1);

    const float* arow = att + ((long)hclamp * NRES + i) * NRES + koff;   // A[h][j]
    const float* zb   = z + (long)i * NRES * CZ + (long)koff * CZ + n0 + frag;

    v8f acc = {};
    for (int j0 = 0; j0 < NRES; j0 += 4) {
        v2f a, b;
        a.x = arow[j0];
        a.y = arow[j0 + 1];
        b.x = zb[(long)j0 * CZ];
        b.y = zb[(long)(j0 + 1) * CZ];
        acc = wmma4(a, b, acc);
    }
    const int mbase = (lane >> 4) << 3;
    #pragma unroll
    for (int r = 0; r < 8; ++r) {
        const int hh = mbase + r;
        if (hh < NHEAD)
            X[(long)i * DCAT + OFF_PAIR + hh * CZ + n0 + frag] = acc[r];
    }
}

// v_out[i, h*16+c] = sum_j att[h,i,j]*v[j, h*16+c]; grid (6, 12), block 128.
__global__ void vout_av(const float* __restrict__ att, const float* __restrict__ vbuf,
                        float* __restrict__ X)
{
    const int lane = threadIdx.x & 31;
    const int w = threadIdx.x >> 5;
    const int h = blockIdx.y;
    const int i0 = (blockIdx.x * 4 + w) << 4;
    const int frag = lane & 15;
    const int koff = (lane >> 4) << 1;

    const float* arow = att + ((long)h * NRES + i0 + frag) * NRES + koff;
    const float* bcol = vbuf + (long)koff * (NHEAD * CH) + h * CH + frag;

    v8f acc = {};
    for (int j0 = 0; j0 < NRES; j0 += 4) {
        v2f a, b;
        a.x = arow[j0]; a.y = arow[j0 + 1];
        b.x = bcol[(long)j0 * (NHEAD * CH)];
        b.y = bcol[(long)(j0 + 1) * (NHEAD * CH)];
        acc = wmma4(a, b, acc);
    }
    const int mbase = (lane >> 4) << 3;
    #pragma unroll
    for (int r = 0; r < 8; ++r)
        X[(long)(i0 + mbase + r) * DCAT + h * CH + frag] = acc[r];
}

// v_att[h,i,p*3+c] = sum_j att[h,i,j]*vt[h,j,p*3+c]; N=24 ragged -> clamp.
// grid (6 itile-groups, 2 ntiles, 12 heads), block 128.
__global__ void vatt_av(const float* __restrict__ att, const float* __restrict__ vt,
                        float* __restrict__ vab)
{
    const int lane = threadIdx.x & 31;
    const int w = threadIdx.x >> 5;
    const int h = blockIdx.z;
    const int i0 = (blockIdx.x * 4 + w) << 4;
    const int n0 = blockIdx.y << 4;
    const int frag = lane & 15;
    const int koff = (lane >> 4) << 1;
    const int nb = n0 + frag;
    const int nbc = nb < 3 * NPV ? nb : (3 * NPV - 1);   // clamped column

    const float* arow = att + ((long)h * NRES + i0 + frag) * NRES + koff;
    const float* bcol = vt + ((long)h * NRES + koff) * (3 * NPV) + nbc;

    v8f acc = {};
    for (int j0 = 0; j0 < NRES; j0 += 4) {
        v2f a, b;
        a.x = arow[j0]; a.y = arow[j0 + 1];
        b.x = bcol[(long)j0 * (3 * NPV)];
        b.y = bcol[(long)(j0 + 1) * (3 * NPV)];
        acc = wmma4(a, b, acc);
    }
    if (nb < 3 * NPV) {
        const int mbase = (lane >> 4) << 3;
        #pragma unroll
        for (int r = 0; r < 8; ++r)
            vab[((long)h * NRES + i0 + mbase + r) * (3 * NPV) + nb] = acc[r];
    }
}

// Inverse warp R^T(x - t), scatter into concat buffer + norms.
__global__ void inv_warp_norm(const float* __restrict__ vab, const float* __restrict__ T,
                              float* __restrict__ X)
{
    int idx = blockIdx.x * blockDim.x + threadIdx.x;   // NHEAD*NPV*NRES
    if (idx >= NHEAD * NPV * NRES) return;
    const int i = idx % NRES;
    const int hp = idx / NRES;
    const int h = hp / NPV, p = hp % NPV;
    const float* Ti = T + (long)i * 16;
    float d[3];
    #pragma unroll
    for (int r = 0; r < 3; ++r)
        d[r] = vab[((long)h * NRES + i) * (3 * NPV) + p * 3 + r] - Ti[r * 4 + 3];
    float o[3];
    #pragma unroll
    for (int c = 0; c < 3; ++c)
        o[c] = Ti[0*4+c]*d[0] + Ti[1*4+c]*d[1] + Ti[2*4+c]*d[2];
    float* Xi = X + (long)i * DCAT;
    #pragma unroll
    for (int c = 0; c < 3; ++c)
        Xi[OFF_VPOUT + (c * NHEAD + h) * NPV + p] = o[c];
    Xi[OFF_NORM + h * NPV + p] = sqrtf(o[0]*o[0] + o[1]*o[1] + o[2]*o[2]);
}

// ---------------------------------------------------------------------------
extern "C" void kernel_launch(void* const* d_in, const int* in_sizes, int n_in,
                              void* d_out, int out_size, void* d_ws, size_t ws_size,
                              hipStream_t stream) {
    const float* s   = (const float*)d_in[0];
    const float* z   = (const float*)d_in[1];
    const float* T   = (const float*)d_in[2];
    const float* Wq  = (const float*)d_in[3];   const float* bq  = (const float*)d_in[4];
    const float* Wk  = (const float*)d_in[5];   const float* bk  = (const float*)d_in[6];
    const float* Wv  = (const float*)d_in[7];   const float* bv  = (const float*)d_in[8];
    const float* Wqp = (const float*)d_in[9];   const float* bqp = (const float*)d_in[10];
    const float* Wkp = (const float*)d_in[11];  const float* bkp = (const float*)d_in[12];
    const float* Wvp = (const float*)d_in[13];  const float* bvp = (const float*)d_in[14];
    const float* Wb  = (const float*)d_in[15];  const float* bb  = (const float*)d_in[16];
    const float* Wo  = (const float*)d_in[17];  const float* bo  = (const float*)d_in[18];
    const float* hw  = (const float*)d_in[19];

    // workspace layout (floats) — total 5,124,096 floats = 20.5 MB
    float* W = (float*)d_ws;
    float* qbuf  = W;               // 384*192
    float* kbuf  = qbuf  + 73728;
    float* vbuf  = kbuf  + 73728;
    float* qpM   = vbuf  + 73728;   // 384*144
    float* kpM   = qpM   + 55296;
    float* vpM   = kpM   + 55296;   // 384*288
    float* tqp   = vpM   + 110592;  // 12*384*12
    float* tkp   = tqp   + 55296;
    float* vt    = tkp   + 55296;   // 12*384*24
    float* biasb = vt    + 110592;  // 147456*12
    float* att   = biasb + 1769472; // 12*384*384
    float* vab   = att   + 1769472; // 12*384*24
    float* X     = vab   + 110592;  // 384*2112

    auto blocks = [](int M, int N) { int t = (M >> 4) * ((N + 15) >> 4); return (t + 3) / 4; };

    // projections
    gemm_f32_wmma<<<blocks(NRES,192),128,0,stream>>>(s, Wq, bq, qbuf, NRES,192,CS, CS,192,192);
    gemm_f32_wmma<<<blocks(NRES,192),128,0,stream>>>(s, Wk, bk, kbuf, NRES,192,CS, CS,192,192);
    gemm_f32_wmma<<<blocks(NRES,192),128,0,stream>>>(s, Wv, bv, vbuf, NRES,192,CS, CS,192,192);
    gemm_f32_wmma<<<blocks(NRES,144),128,0,stream>>>(s, Wqp, bqp, qpM, NRES,144,CS, CS,144,144);
    gemm_f32_wmma<<<blocks(NRES,144),128,0,stream>>>(s, Wkp, bkp, kpM, NRES,144,CS, CS,144,144);
    gemm_f32_wmma<<<blocks(NRES,288),128,0,stream>>>(s, Wvp, bvp, vpM, NRES,288,CS, CS,288,288);
    // bias = z @ Wb + bb   (M = 384*384, N = 12 ragged)
    gemm_f32_wmma<<<blocks(NRES*NRES,12),128,0,stream>>>(z, Wb, bb, biasb,
                                                         NRES*NRES,12,CZ, CZ,12,12);
    // frame-warp points
    warp_points<<<(NHEAD*NQP*NRES)/256,256,0,stream>>>(qpM, T, tqp, NQP);
    warp_points<<<(NHEAD*NQP*NRES)/256,256,0,stream>>>(kpM, T, tkp, NQP);
    warp_points<<<(NHEAD*NPV*NRES)/256,256,0,stream>>>(vpM, T, vt,  NPV);
    // logits + softmax
    ipa_logits<<<dim3(24,6,NHEAD),128,0,stream>>>(qbuf, kbuf, biasb, tqp, tkp, hw, att);
    softmax_rows<<<NHEAD*NRES,32,0,stream>>>(att);
    // attention-weighted contractions
    pairwise_av<<<dim3(NRES,2),128,0,stream>>>(att, z, X);
    vout_av<<<dim3(6,NHEAD),128,0,stream>>>(att, vbuf, X);
    vatt_av<<<dim3(6,2,NHEAD),128,0,stream>>>(att, vt, vab);
    inv_warp_norm<<<(NHEAD*NPV*NRES)/256,256,0,stream>>>(vab, T, X);
    // final projection
    gemm_f32_wmma<<<blocks(NRES,CS),128,0,stream>>>(X, Wo, bo, (float*)d_out,
                                                    NRES,CS,DCAT, DCAT,CS,CS);
    (void)in_sizes; (void)n_in; (void)out_size; (void)ws_size;
}